// TripletLoss_13675175871121
// MI455X (gfx1250) — compile-verified
//
#include <hip/hip_runtime.h>
#include <hip/hip_bf16.h>

#define N 512
#define D 256
#define MARGIN 0.5f
#define TRIP_EPS 1e-16f

typedef __attribute__((ext_vector_type(2))) float v2f;
typedef __attribute__((ext_vector_type(8))) float v8f;

// ---------------------------------------------------------------------------
// Kernel 1: per-row squared norms. One 256-thread block per row (D=256).
// ---------------------------------------------------------------------------
__global__ void row_sqnorm_kernel(const float* __restrict__ E,
                                  float* __restrict__ sqn) {
    __shared__ float red[256];
    const int row = blockIdx.x;
    const int t   = threadIdx.x;
    float v = E[row * D + t];
    red[t] = v * v;
    __syncthreads();
    for (int s = 128; s > 0; s >>= 1) {
        if (t < s) red[t] += red[t + s];
        __syncthreads();
    }
    if (t == 0) sqn[row] = red[0];
}

// ---------------------------------------------------------------------------
// Kernel 2: Gram = E * E^T via V_WMMA_F32_16X16X4_F32, then euclidean dist.
// One wave32 per 16x16 output tile; 8 waves (8 i-tiles) per block.
// A (16x4 f32) ISA layout: lane m (0..15)/half h: VGPR0=K[2h], VGPR1=K[2h+1].
// B (4x16) mirrors it with N along lanes, so for E*E^T both operands load
// with the identical float2 pattern from their respective row blocks.
// C/D layout: VGPR r -> row = r + 8*half, col = lane&15.
// ---------------------------------------------------------------------------
__global__ void gram_dist_kernel(const float* __restrict__ E,
                                 const float* __restrict__ sqn,
                                 float* __restrict__ pd) {
    const int lane  = threadIdx.x;            // 0..31
    const int half  = lane >> 4;              // 0/1
    const int m     = lane & 15;              // 0..15
    const int iTile = blockIdx.y * 8 + threadIdx.y;  // 0..31
    const int jTile = blockIdx.x;                    // 0..31

    const float* __restrict__ pa = E + (iTile * 16 + m) * D + 2 * half;
    const float* __restrict__ pb = E + (jTile * 16 + m) * D + 2 * half;

    v8f c = {};
#pragma unroll 4
    for (int k = 0; k < D; k += 4) {
        v2f a = *(const v2f*)(pa + k);
        v2f b = *(const v2f*)(pb + k);
        // 8 args: (neg_a, A, neg_b, B, c_mod, C, reuse_a, reuse_b)
        c = __builtin_amdgcn_wmma_f32_16x16x4_f32(
            false, a, false, b, (short)0, c, false, false);
    }

#pragma unroll
    for (int r = 0; r < 8; ++r) {
        const int row = iTile * 16 + r + 8 * half;
        const int col = jTile * 16 + m;
        float d = sqn[row] + sqn[col] - 2.0f * c[r];
        d = fmaxf(d, 0.0f);
        pd[row * N + col] = (d > 0.0f) ? __builtin_sqrtf(d) : 0.0f;
    }
}

// ---------------------------------------------------------------------------
// Kernel 3: per-anchor triplet reduction. One 256-thread block per anchor.
// mask(a,p,n) = (p!=a) & (lab[p]==lab[a]) & (lab[n]!=lab[a])   (implies
// a!=n and p!=n automatically). Deterministic tree reduction, no atomics.
// partials: [0..N) loss sums, [N..2N) positive counts, [2N..3N) valid counts.
// ---------------------------------------------------------------------------
__global__ void triplet_kernel(const int* __restrict__ labels,
                               const float* __restrict__ pd,
                               float* __restrict__ partials) {
    __shared__ float sd[N];
    __shared__ int   sl[N];
    __shared__ float rsum[256];
    __shared__ float rpos[256];
    __shared__ float rval[256];

    const int a = blockIdx.x;
    const int t = threadIdx.x;

    for (int i = t; i < N; i += 256) {
        sd[i] = pd[a * N + i];
        sl[i] = labels[i];
    }
    __syncthreads();

    const int la = sl[a];
    float sum = 0.0f, pos = 0.0f, val = 0.0f;

    for (int p = 0; p < N; ++p) {
        if (p == a || sl[p] != la) continue;
        const float dap = sd[p];
#pragma unroll
        for (int nn = 0; nn < 2; ++nn) {
            const int n = t + nn * 256;
            if (sl[n] != la) {
                val += 1.0f;
                const float v = dap - sd[n] + MARGIN;
                if (v > 0.0f) {
                    sum += v;
                    if (v > TRIP_EPS) pos += 1.0f;
                }
            }
        }
    }

    rsum[t] = sum; rpos[t] = pos; rval[t] = val;
    __syncthreads();
    for (int s = 128; s > 0; s >>= 1) {
        if (t < s) {
            rsum[t] += rsum[t + s];
            rpos[t] += rpos[t + s];
            rval[t] += rval[t + s];
        }
        __syncthreads();
    }
    if (t == 0) {
        partials[a]         = rsum[0];
        partials[N + a]     = rpos[0];
        partials[2 * N + a] = rval[0];
    }
}

// ---------------------------------------------------------------------------
// Kernel 4: finalize. Single 256-thread block reduces 512 partials each.
// out[0] = loss, out[1] = fraction_positive.
// ---------------------------------------------------------------------------
__global__ void finalize_kernel(const float* __restrict__ partials,
                                float* __restrict__ out) {
    __shared__ float rsum[256];
    __shared__ float rpos[256];
    __shared__ float rval[256];
    const int t = threadIdx.x;
    rsum[t] = partials[t]         + partials[t + 256];
    rpos[t] = partials[N + t]     + partials[N + t + 256];
    rval[t] = partials[2 * N + t] + partials[2 * N + t + 256];
    __syncthreads();
    for (int s = 128; s > 0; s >>= 1) {
        if (t < s) {
            rsum[t] += rsum[t + s];
            rpos[t] += rpos[t + s];
            rval[t] += rval[t + s];
        }
        __syncthreads();
    }
    if (t == 0) {
        out[0] = rsum[0] / (rpos[0] + TRIP_EPS);
        out[1] = rpos[0] / (rval[0] + TRIP_EPS);
    }
}

// ---------------------------------------------------------------------------
extern "C" void kernel_launch(void* const* d_in, const int* in_sizes, int n_in,
                              void* d_out, int out_size, void* d_ws, size_t ws_size,
                              hipStream_t stream) {
    const int*   labels = (const int*)d_in[0];        // harness: integer -> int*
    const float* E      = (const float*)d_in[1];      // (512, 256) fp32
    float*       out    = (float*)d_out;              // 2 floats

    // workspace layout: sqn[N] | pd[N*N] | partials[3*N]
    float* sqn      = (float*)d_ws;
    float* pd       = sqn + N;
    float* partials = pd + N * N;

    row_sqnorm_kernel<<<dim3(N), dim3(256), 0, stream>>>(E, sqn);

    gram_dist_kernel<<<dim3(N / 16, (N / 16) / 8), dim3(32, 8), 0, stream>>>(E, sqn, pd);

    triplet_kernel<<<dim3(N), dim3(256), 0, stream>>>(labels, pd, partials);

    finalize_kernel<<<dim3(1), dim3(256), 0, stream>>>(partials, out);
}